// NodeEmbeddingModule_64415919505525
// MI455X (gfx1250) — compile-verified
//
#include <hip/hip_runtime.h>
#include <stdint.h>

typedef __attribute__((ext_vector_type(2))) float v2f;
typedef __attribute__((ext_vector_type(8))) float v8f;

// ---------------------------------------------------------------------------
// Degree / normalization precompute (shared by both GCN layers)
// ---------------------------------------------------------------------------
__global__ void k_init_deg(float* __restrict__ deg, int n) {
  int i = blockIdx.x * blockDim.x + threadIdx.x;
  if (i < n) deg[i] = 1.0f;  // self-loop weight 1
}

__global__ void k_accum_deg(float* __restrict__ deg,
                            const long long* __restrict__ dst,
                            const float* __restrict__ ew, int E) {
  int e = blockIdx.x * blockDim.x + threadIdx.x;
  if (e < E) atomicAdd(&deg[(int)dst[e]], ew[e]);
}

__global__ void k_make_dinv(float* __restrict__ deg, int n) {
  int i = blockIdx.x * blockDim.x + threadIdx.x;
  if (i < n) {
    float d = deg[i];
    deg[i] = (d > 0.0f) ? rsqrtf(d) : 0.0f;
  }
}

// ---------------------------------------------------------------------------
// Repack B[K,N] (K even) into K-pair-interleaved form so one lane's WMMA B
// fragment {B[k][n], B[k+1][n]} is a single contiguous 8-byte load:
//   Bp[(k/2)*N + n] = { B[k][n], B[k+1][n] }   (stored as v2f)
// ---------------------------------------------------------------------------
__global__ void k_pack_pairs(const float* __restrict__ B, float* __restrict__ Bp,
                             int K, int N) {
  int idx = blockIdx.x * blockDim.x + threadIdx.x;  // over (K/2)*N
  int total = (K >> 1) * N;
  if (idx >= total) return;
  int p = idx / N;          // k-pair index, k = 2p
  int col = idx - p * N;
  int k = p << 1;
  Bp[2 * idx + 0] = B[(size_t)k * N + col];
  Bp[2 * idx + 1] = B[(size_t)(k + 1) * N + col];
}

// ---------------------------------------------------------------------------
// Dense GEMM  C[M,N] = A[M,K] * Bp  using V_WMMA_F32_16X16X4_F32.
// NT = N/16 is a compile-time constant -> straight-line unrolled chunk body
// (no per-tile branches), letting loads batch ahead of the 8 WMMAs.
//
// 32-bit A 16x4 layout:  lanes 0-15 -> M=lane,    K={k0,k0+1} in v[0],v[1]
//                        lanes 16-31 -> M=lane-16, K={k0+2,k0+3}
// B 4x16 mirrors A with N in place of M; fragment read from packed Bp.
// C/D: VGPR i holds row i (+8 for upper lane half), N = lane&15.
// ---------------------------------------------------------------------------
template <int NT>
__global__ void k_gemm_wmma_f32(const float* __restrict__ A,
                                const v2f* __restrict__ Bp,
                                float* __restrict__ C,
                                int M, int K, int N) {
  int wave = (int)((blockIdx.x * blockDim.x + threadIdx.x) >> 5);
  int lane = (int)(threadIdx.x & 31);
  int half = lane >> 4;
  int l16  = lane & 15;
  int mtiles = (M + 15) >> 4;
  if (wave >= mtiles) return;  // wave-uniform: EXEC stays all-ones for WMMA

  int row  = wave * 16 + l16;
  int rowc = (row < M) ? row : (M - 1);  // clamp loads (no divergence)

  v8f acc[NT];
#pragma unroll
  for (int t = 0; t < NT; ++t)
    acc[t] = (v8f){0.f, 0.f, 0.f, 0.f, 0.f, 0.f, 0.f, 0.f};

  const float* Arow = A + (size_t)rowc * K;
#pragma unroll 2
  for (int k0 = 0; k0 < K; k0 += 4) {
    int kk = k0 + 2 * half;                       // even
    v2f a = *(const v2f*)(Arow + kk);             // A[row][kk], A[row][kk+1]
    const v2f* Bk = Bp + (size_t)(kk >> 1) * N + l16;
    v2f b[NT];
#pragma unroll
    for (int t = 0; t < NT; ++t) b[t] = Bk[t * 16];
#pragma unroll
    for (int t = 0; t < NT; ++t) {
      acc[t] = __builtin_amdgcn_wmma_f32_16x16x4_f32(
          /*neg_a=*/false, a, /*neg_b=*/false, b[t],
          /*c_mod=*/(short)0, acc[t], /*reuse_a=*/false, /*reuse_b=*/false);
    }
  }

#pragma unroll
  for (int t = 0; t < NT; ++t) {
    int col = t * 16 + l16;
#pragma unroll
    for (int i = 0; i < 8; ++i) {
      int r = wave * 16 + i + 8 * half;
      if (r < M) C[(size_t)r * N + col] = acc[t][i];
    }
  }
}

// ---------------------------------------------------------------------------
// Aggregation: acc[i] = h[i] * dinv[i]^2  (self loop, norm = dinv*1*dinv)
// ---------------------------------------------------------------------------
__global__ void k_agg_self(const float* __restrict__ H,
                           const float* __restrict__ dinv,
                           float* __restrict__ Acc, int n, int C) {
  int idx = blockIdx.x * blockDim.x + threadIdx.x;  // over n * C/4
  int c4n = C >> 2;
  int total = n * c4n;
  if (idx >= total) return;
  int i  = idx / c4n;
  int c4 = idx - i * c4n;
  float s = dinv[i];
  s = s * s;
  float4 v = ((const float4*)(H + (size_t)i * C))[c4];
  v.x *= s; v.y *= s; v.z *= s; v.w *= s;
  ((float4*)(Acc + (size_t)i * C))[c4] = v;
}

// One wave per edge: lane handles 4 channels (C=128 -> 32 lanes x float4).
__global__ void k_agg_edges(const float* __restrict__ H,
                            const float* __restrict__ dinv,
                            const long long* __restrict__ src,
                            const long long* __restrict__ dst,
                            const float* __restrict__ ew,
                            float* __restrict__ Acc, int E, int C) {
  int wave = (int)((blockIdx.x * blockDim.x + threadIdx.x) >> 5);
  int lane = (int)(threadIdx.x & 31);
  if (wave >= E) return;
  int s = (int)src[wave];
  int d = (int)dst[wave];
  float norm = dinv[s] * ew[wave] * dinv[d];
  const float* hs = H + (size_t)s * C;
  float* ad = Acc + (size_t)d * C;
  for (int c = lane * 4; c < C; c += 128) {
    float4 v = *(const float4*)(hs + c);
    atomicAdd(ad + c + 0, v.x * norm);
    atomicAdd(ad + c + 1, v.y * norm);
    atomicAdd(ad + c + 2, v.z * norm);
    atomicAdd(ad + c + 3, v.w * norm);
  }
}

__global__ void k_bias_relu(const float* __restrict__ Acc,
                            const float* __restrict__ bias,
                            float* __restrict__ O, int n, int C) {
  int idx = blockIdx.x * blockDim.x + threadIdx.x;
  int total = n * C;
  if (idx >= total) return;
  int c = idx & (C - 1);  // C is a power of two (128)
  float v = Acc[idx] + bias[c];
  O[idx] = v > 0.0f ? v : 0.0f;
}

// ---------------------------------------------------------------------------
// Launcher
// ---------------------------------------------------------------------------
extern "C" void kernel_launch(void* const* d_in, const int* in_sizes, int n_in,
                              void* d_out, int out_size, void* d_ws, size_t ws_size,
                              hipStream_t stream) {
  const float*     x   = (const float*)d_in[0];
  const long long* ei  = (const long long*)d_in[1];  // int64 [2, E]
  const float*     ew  = (const float*)d_in[2];
  const float*     W1  = (const float*)d_in[3];
  const float*     b1  = (const float*)d_in[4];
  const float*     W2  = (const float*)d_in[5];
  const float*     b2  = (const float*)d_in[6];
  float*           out = (float*)d_out;

  const int chid = in_sizes[4];           // 128
  const int cout = in_sizes[6];           // 128
  const int cin  = in_sizes[3] / chid;    // 256
  const int n    = in_sizes[0] / cin;     // 50000
  const int E    = in_sizes[2];           // 800000
  const long long* src = ei;
  const long long* dst = ei + E;

  // Workspace layout: dinv | H (n*128) | Acc (n*128) | Bp1 | Bp2
  char* ws = (char*)d_ws;
  float* dinv = (float*)ws;
  size_t off  = (((size_t)n * sizeof(float)) + 255) & ~(size_t)255;
  size_t fbuf = (((size_t)n * chid * sizeof(float)) + 255) & ~(size_t)255;
  float* H   = (float*)(ws + off);
  float* Acc = (float*)(ws + off + fbuf);
  float* Bp1 = (float*)(ws + off + 2 * fbuf);
  size_t b1sz = (((size_t)cin * chid * sizeof(float)) + 255) & ~(size_t)255;
  float* Bp2 = (float*)(ws + off + 2 * fbuf + b1sz);

  const int T = 256;
  // --- normalization (shared by both layers) ---
  k_init_deg <<<(n + T - 1) / T, T, 0, stream>>>(dinv, n);
  k_accum_deg<<<(E + T - 1) / T, T, 0, stream>>>(dinv, dst, ew, E);
  k_make_dinv<<<(n + T - 1) / T, T, 0, stream>>>(dinv, n);

  // --- weight repack for single-b64 B fragments ---
  k_pack_pairs<<<(((cin >> 1) * chid) + T - 1) / T, T, 0, stream>>>(W1, Bp1, cin, chid);
  k_pack_pairs<<<(((chid >> 1) * cout) + T - 1) / T, T, 0, stream>>>(W2, Bp2, chid, cout);

  // --- layer 1: H = x@W1 ; Acc = scatter(H) ; H = relu(Acc + b1) ---
  {
    int waves  = (n + 15) / 16;
    int blocks = (waves + 3) / 4;  // 4 waves per 128-thread block
    k_gemm_wmma_f32<8><<<blocks, 128, 0, stream>>>(x, (const v2f*)Bp1, H, n, cin, chid);
  }
  k_agg_self <<<((n * (chid >> 2)) + T - 1) / T, T, 0, stream>>>(H, dinv, Acc, n, chid);
  k_agg_edges<<<(E + 7) / 8, 256, 0, stream>>>(H, dinv, src, dst, ew, Acc, E, chid);
  k_bias_relu<<<((n * chid) + T - 1) / T, T, 0, stream>>>(Acc, b1, H, n, chid);

  // --- layer 2: Acc = H@W2 ; out = scatter(Acc) ; out = relu(out + b2) ---
  {
    int waves  = (n + 15) / 16;
    int blocks = (waves + 3) / 4;
    k_gemm_wmma_f32<8><<<blocks, 128, 0, stream>>>(H, (const v2f*)Bp2, Acc, n, chid, cout);
  }
  k_agg_self <<<((n * (cout >> 2)) + T - 1) / T, T, 0, stream>>>(Acc, dinv, out, n, cout);
  k_agg_edges<<<(E + 7) / 8, 256, 0, stream>>>(Acc, dinv, src, dst, ew, out, E, cout);
  k_bias_relu<<<((n * cout) + T - 1) / T, T, 0, stream>>>(out, b2, out, n, cout);
}